// MobiusFlow4ND_78572131713171
// MI455X (gfx1250) — compile-verified
//
#include <hip/hip_runtime.h>
#include <cstdint>

#define KK 64
#define CND (4 * KK)   // 256 floats of conditions per (b,n)
#define WPB 8          // (b,n) items (waves) per block
#define BLOCK (WPB * 32)
#define TWO_PI 6.28318530717958647692f

typedef int v4i __attribute__((__vector_size__(4 * sizeof(int))));
typedef __attribute__((address_space(1))) v4i gbl_v4i;
typedef __attribute__((address_space(3))) v4i lds_v4i;

// ---- gfx1250 async global->LDS copy (ASYNCcnt path) ----------------------
__device__ __forceinline__ void async_b128(const float* g, const float* l) {
#if __has_builtin(__builtin_amdgcn_global_load_async_to_lds_b128)
  __builtin_amdgcn_global_load_async_to_lds_b128(
      (gbl_v4i*)(uintptr_t)g, (lds_v4i*)(uintptr_t)l, 0, 0);
#else
  unsigned loff = (unsigned)(uintptr_t)l;  // low 32 bits of flat LDS addr == LDS offset
  asm volatile("global_load_async_to_lds_b128 %0, %1, off"
               :: "v"(loff), "v"(g) : "memory");
#endif
}

__device__ __forceinline__ void wait_async0() {
#if __has_builtin(__builtin_amdgcn_s_wait_asynccnt)
  __builtin_amdgcn_s_wait_asynccnt(0);
#else
  asm volatile("s_wait_asynccnt 0" ::: "memory");
#endif
  asm volatile("" ::: "memory");  // compiler barrier: keep LDS reads after wait
}

// --------------------------------------------------------------------------
__global__ __launch_bounds__(BLOCK) void mobius_main(
    const float* __restrict__ rot,        // (BN, 3, 3)
    const float* __restrict__ cond,       // (BN, 256)
    const long long* __restrict__ perm,   // (3,)
    float* __restrict__ out_rot,          // (BN, 3, 3)
    float* __restrict__ ldj_ws,           // (BN,)
    int bn_total) {
  __shared__ float scond[WPB * CND];      // 8 KB

  const int wave = threadIdx.x >> 5;
  const int lane = threadIdx.x & 31;
  int bn = blockIdx.x * WPB + wave;
  if (bn >= bn_total) return;             // wave-uniform guard
  // bn is wave-uniform: tell the compiler so rotation/output addressing
  // and the per-item frame math can live in SGPRs / scalar loads.
  bn = __builtin_amdgcn_readfirstlane(bn);

  // --- stage this wave's 256 conditions floats into LDS (2 x b128 / lane) --
  const float* g0 = cond + (size_t)bn * CND + lane * 4;
  const float* l0 = &scond[wave * CND + lane * 4];
  async_b128(g0, l0);
  async_b128(g0 + 128, l0 + 128);

  const int p0 = (int)perm[0], p1 = (int)perm[1], p2 = (int)perm[2];

  // --- per-item frame (overlapped with the async DMA) ----------------------
  const float* R = rot + (size_t)bn * 9;
  const float x0 = R[0 * 3 + p0], x1 = R[1 * 3 + p0], x2 = R[2 * 3 + p0];
  const float y0 = R[0 * 3 + p1], y1 = R[1 * 3 + p1], y2 = R[2 * 3 + p1];

  // r = -x / |x|
  const float inx = rsqrtf(x0 * x0 + x1 * x1 + x2 * x2);
  const float r0 = -x0 * inx, r1 = -x1 * inx, r2 = -x2 * inx;
  // v = normalize(cross(y, r))
  float v0 = y1 * r2 - y2 * r1;
  float v1 = y2 * r0 - y0 * r2;
  float v2 = y0 * r1 - y1 * r0;
  const float ivn = rsqrtf(v0 * v0 + v1 * v1 + v2 * v2);
  v0 *= ivn; v1 *= ivn; v2 *= ivn;
  // theta and dz/dtheta = -sin(theta) r + cos(theta) v
  const float theta = atan2f(x0 * v0 + x1 * v1 + x2 * v2,
                             x0 * r0 + x1 * r1 + x2 * r2);
  const float st = sinf(theta), ct = cosf(theta);
  const float d0 = -st * r0 + ct * v0;
  const float d1 = -st * r1 + ct * v1;
  const float d2 = -st * r2 + ct * v2;

  wait_async0();

  // --- K-loop: lane handles k = lane and lane+32 ---------------------------
  const float* sc = &scond[wave * CND];
  float Sw = 0.f, Sang = 0.f, Sd = 0.f;
#pragma unroll
  for (int kk = 0; kk < 2; ++kk) {
    const int k = lane + kk * 32;
    const float wr = sc[k];
    const float sp = fmaxf(wr, 0.0f) + log1pf(expf(-fabsf(wr)));  // softplus

    float w0 = sc[KK + 3 * k + 0];
    float w1 = sc[KK + 3 * k + 1];
    float w2 = sc[KK + 3 * k + 2];
    // project: w -= y (y.w)
    const float yw = y0 * w0 + y1 * w1 + y2 * w2;
    w0 -= y0 * yw; w1 -= y1 * yw; w2 -= y2 * yw;
    // scale: w *= 0.7/(1+|w|)
    const float wn = sqrtf(w0 * w0 + w1 * w1 + w2 * w2);
    const float s = 0.7f / (1.0f + wn);
    w0 *= s; w1 *= s; w2 *= s;
    const float wn2 = w0 * w0 + w1 * w1 + w2 * w2;
    // zw = x - w
    const float q0 = x0 - w0, q1 = x1 - w1, q2 = x2 - w2;
    const float zw2 = q0 * q0 + q1 * q1 + q2 * q2;
    const float c = (1.0f - wn2) / zw2;
    // h_z = c*zw - w
    const float h0 = c * q0 - w0, h1 = c * q1 - w1, h2 = c * q2 - w2;
    const float rad = atan2f(h0 * v0 + h1 * v1 + h2 * v2,
                             h0 * r0 + h1 * r1 + h2 * r2);
    const float angk = (rad >= 0.0f) ? rad : rad + TWO_PI;
    // ||dh/dtheta|| = c * || dzt - 2 u (u.dzt) ||,  u = zw/|zw|
    const float izw = rsqrtf(zw2);
    const float u0 = q0 * izw, u1 = q1 * izw, u2 = q2 * izw;
    const float ud = u0 * d0 + u1 * d1 + u2 * d2;
    const float m0 = d0 - 2.0f * ud * u0;
    const float m1 = d1 - 2.0f * ud * u1;
    const float m2 = d2 - 2.0f * ud * u2;
    const float dn = c * sqrtf(m0 * m0 + m1 * m1 + m2 * m2);

    Sw += sp; Sang += sp * angk; Sd += sp * dn;
  }

  // --- wave32 reductions ---------------------------------------------------
#pragma unroll
  for (int off = 16; off >= 1; off >>= 1) {
    Sw   += __shfl_xor(Sw, off, 32);
    Sang += __shfl_xor(Sang, off, 32);
    Sd   += __shfl_xor(Sd, off, 32);
  }

  if (lane == 0) {
    const float ang = Sang / Sw;          // normalized weighted angle
    const float ca = cosf(ang), sa = sinf(ang);
    const float t0 = r0 * ca + v0 * sa;
    const float t1 = r1 * ca + v1 * sa;
    const float t2 = r2 * ca + v2 * sa;
    float z0, z1, z2;
    const int dp = p1 - p0;
    if (dp == 1 || dp == -2) {            // tz = cross(tx, y)
      z0 = t1 * y2 - t2 * y1;
      z1 = t2 * y0 - t0 * y2;
      z2 = t0 * y1 - t1 * y0;
    } else {                              // tz = cross(y, tx)
      z0 = y1 * t2 - y2 * t1;
      z1 = y2 * t0 - y0 * t2;
      z2 = y0 * t1 - y1 * t0;
    }
    const float izn = rsqrtf(z0 * z0 + z1 * z1 + z2 * z2);
    z0 *= izn; z1 *= izn; z2 *= izn;

    float* O = out_rot + (size_t)bn * 9;
    O[0 * 3 + p0] = t0; O[1 * 3 + p0] = t1; O[2 * 3 + p0] = t2;
    O[0 * 3 + p1] = y0; O[1 * 3 + p1] = y1; O[2 * 3 + p1] = y2;
    O[0 * 3 + p2] = z0; O[1 * 3 + p2] = z1; O[2 * 3 + p2] = z2;

    ldj_ws[bn] = logf(Sd / Sw);           // log det J for this (b,n)
  }
}

// Deterministic per-batch sum of ldj over N (no float atomics).
__global__ void ldj_reduce(const float* __restrict__ ldj_ws,
                           float* __restrict__ out, int Bv, int Nv) {
  const int b = blockIdx.x * blockDim.x + threadIdx.x;
  if (b < Bv) {
    const float* p = ldj_ws + (size_t)b * Nv;
    float s = 0.f;
#pragma unroll 7
    for (int n = 0; n < Nv; ++n) s += p[n];
    out[b] = s;
  }
}

extern "C" void kernel_launch(void* const* d_in, const int* in_sizes, int n_in,
                              void* d_out, int out_size, void* d_ws, size_t ws_size,
                              hipStream_t stream) {
  const float*      rot  = (const float*)d_in[0];
  const float*      cond = (const float*)d_in[1];
  const long long*  perm = (const long long*)d_in[2];

  const int BN   = in_sizes[1] / CND;     // 86016
  const int trot = BN * 9;                // 774144 floats of trotation
  const int Bv   = out_size - trot;       // 4096
  const int Nv   = BN / Bv;               // 21

  float* out_rot = (float*)d_out;
  float* out_ldj = (float*)d_out + trot;
  float* ldj_ws  = (float*)d_ws;          // BN floats of scratch

  const int grid1 = (BN + WPB - 1) / WPB;
  mobius_main<<<grid1, BLOCK, 0, stream>>>(rot, cond, perm, out_rot, ldj_ws, BN);

  const int grid2 = (Bv + 255) / 256;
  ldj_reduce<<<grid2, 256, 0, stream>>>(ldj_ws, out_ldj, Bv, Nv);
}